// Autoencoder_65223373357102
// MI455X (gfx1250) — compile-verified
//
#include <hip/hip_runtime.h>
#include <hip/hip_bf16.h>

typedef __attribute__((ext_vector_type(2))) float v2f;
typedef __attribute__((ext_vector_type(8))) float v8f;

#define BB   64
#define VV   5023
#define VM   3500
#define LAT  556
#define DIN  150528
#define NS   400
#define NV3  (VV*3)        // 15069
#define NP   560           // padded LAT for partials
#define KSPLIT 64
#define KC   (DIN/KSPLIT)  // 2352
#define NCH  (KC/16)       // 147 chunks of K=16 per block
#define VOUT 7079          // 3500+3500+68+11

// ---------------- helpers ----------------
__device__ inline void axis_angle_to_R(float a0, float a1, float a2, float* R) {
    float th  = sqrtf(a0*a0 + a1*a1 + a2*a2);
    float inv = 1.0f / fmaxf(th, 1e-8f);
    float u0 = a0*inv, u1 = a1*inv, u2 = a2*inv;
    float s = sinf(th), c = cosf(th), t = 1.0f - c;
    float uu = u0*u0 + u1*u1 + u2*u2;           // K^2 = u u^T - (u.u) I
    R[0] = 1.0f + t*(u0*u0 - uu); R[1] = -s*u2 + t*u0*u1;       R[2] =  s*u1 + t*u0*u2;
    R[3] =  s*u2 + t*u0*u1;       R[4] = 1.0f + t*(u1*u1 - uu); R[5] = -s*u0 + t*u1*u2;
    R[6] = -s*u1 + t*u0*u2;       R[7] =  s*u0 + t*u1*u2;       R[8] = 1.0f + t*(u2*u2 - uu);
}

// out_i = sum_l (x_l - vm_l) * R[l,i] * scale + tr_i   (einsum 'bvl,bli->bvi')
__device__ inline void transform_point(const float* x, const float* vm, const float* R,
                                       float scale, const float* tr, float* o) {
    float d0 = x[0]-vm[0], d1 = x[1]-vm[1], d2 = x[2]-vm[2];
    o[0] = (d0*R[0] + d1*R[3] + d2*R[6])*scale + tr[0];
    o[1] = (d0*R[1] + d1*R[4] + d2*R[7])*scale + tr[1];
    o[2] = (d0*R[2] + d1*R[5] + d2*R[8])*scale + tr[2];
}

__device__ inline float triple(const float* a, const float* b, const float* c) {
    return a[0]*(b[1]*c[2]-b[2]*c[1]) - a[1]*(b[0]*c[2]-b[2]*c[0]) + a[2]*(b[0]*c[1]-b[1]*c[0]);
}

// generic (shared) pointer -> wave-relative LDS byte offset (low 32 bits of aperture addr)
__device__ inline unsigned lds_off(const void* p) {
    return (unsigned)(uintptr_t)p;
}

// ---------------- K0: per-shape-component channel means ----------------
__global__ void means_kernel(const float* __restrict__ basis, const float* __restrict__ tmpl,
                             float* __restrict__ bmean) {
    __shared__ float red[256*3];
    int s = blockIdx.x, t = threadIdx.x;
    const float* src = (s < NS) ? basis + (long)s*NV3 : tmpl;
    float a0 = 0.f, a1 = 0.f, a2 = 0.f;
    for (int v = t; v < VV; v += 256) {
        a0 += src[v*3]; a1 += src[v*3+1]; a2 += src[v*3+2];
    }
    red[t] = a0; red[256+t] = a1; red[512+t] = a2;
    __syncthreads();
    for (int off = 128; off > 0; off >>= 1) {
        if (t < off) { red[t] += red[t+off]; red[256+t] += red[256+t+off]; red[512+t] += red[512+t+off]; }
        __syncthreads();
    }
    if (t == 0) {
        float inv = 1.0f / (float)VV;
        float* dst = bmean + ((s < NS) ? s*3 : NS*3);
        dst[0] = red[0]*inv; dst[1] = red[256]*inv; dst[2] = red[512]*inv;
    }
}

// ---------------- K1: latent GEMM, fp32 WMMA 16x16x4, split-K ----------------
// grid (35 n-tiles, 64 k-chunks), block 128 (4 waves; wave = m-tile).
// B (enc_W) 16x16 chunks are staged once per block into LDS with
// global_load_async_to_lds_b128 (double-buffered, ASYNCcnt-synchronized).
__global__ void gemm1_wmma(const float* __restrict__ X, const float* __restrict__ W,
                           float* __restrict__ part) {
    __shared__ float lb[2][16][16];   // [buf][k][n] chunk of W
    int tid  = threadIdx.x;
    int wave = tid >> 5;              // m-tile 0..3
    int lane = tid & 31;
    int half = lane >> 4, lid = lane & 15;
    int n_tile = blockIdx.x, kc = blockIdx.y;
    int n0 = n_tile*16;
    int m0 = wave*16;
    int kbeg = kc*KC;

    // staging lanes: threads 0..63 -> row r = tid>>2, 4-col group q = tid&3
    int sr = tid >> 2;
    int sq = (tid & 3) * 4;
    int gcol = n0 + sq; if (gcol > LAT - 4) gcol = LAT - 4;   // clamp; garbage cols discarded later
    const float* gsrc0 = W + (long)(kbeg + sr)*LAT + gcol;

    if (tid < 64) {
        unsigned lo = lds_off(&lb[0][sr][sq]);
        asm volatile("global_load_async_to_lds_b128 %0, %1, off"
                     :: "v"(lo), "v"(gsrc0) : "memory");
    }

    const float* ap = X + (long)(m0 + lid)*DIN + kbeg + 2*half;
    v8f acc = {};
    for (int c = 0; c < NCH; ++c) {
        int cur = c & 1;
        asm volatile("s_wait_asynccnt 0x0" ::: "memory");
        __syncthreads();                       // staged chunk c visible; prev buffer free
        if (c + 1 < NCH && tid < 64) {
            const float* gsrc = gsrc0 + (long)(c + 1)*16*LAT;
            unsigned lo = lds_off(&lb[cur ^ 1][sr][sq]);
            asm volatile("global_load_async_to_lds_b128 %0, %1, off"
                         :: "v"(lo), "v"(gsrc) : "memory");
        }
        #pragma unroll
        for (int u = 0; u < 4; ++u) {
            int kl = u*4 + 2*half;
            v2f a = *(const v2f*)(ap + u*4);
            v2f bf; bf.x = lb[cur][kl][lid]; bf.y = lb[cur][kl + 1][lid];
            acc = __builtin_amdgcn_wmma_f32_16x16x4_f32(false, a, false, bf,
                                                        (short)0, acc, false, false);
        }
        ap += 16;
    }

    // C/D layout: acc[r] = D[m0 + 8*half + r][n0 + lid]
    float* dst = part + ((long)kc*BB + m0 + 8*half)*NP + n0 + lid;
    #pragma unroll
    for (int r = 0; r < 8; ++r) dst[(long)r*NP] = acc[r];
}

// ---------------- K2: split-K reduce + bias ----------------
__global__ void reduce_latent(const float* __restrict__ part, const float* __restrict__ bias,
                              float* __restrict__ latent) {
    int i = blockIdx.x*blockDim.x + threadIdx.x;
    if (i >= BB*LAT) return;
    int m = i / LAT, n = i % LAT;
    float s = bias[n];
    #pragma unroll 4
    for (int kc = 0; kc < KSPLIT; ++kc) s += part[((long)kc*BB + m)*NP + n];
    latent[i] = s;
}

// ---------------- K3: shape-basis GEMM, fp32 WMMA 16x16x4 ----------------
// vert_raw[b][n] = v_template[n] + sum_s latent[b][s] * basis[s][n], n over 15069
__global__ void gemm2_wmma(const float* __restrict__ A, const float* __restrict__ Bm,
                           const float* __restrict__ tmpl, float* __restrict__ vert) {
    int wave = threadIdx.x >> 5;
    int lane = threadIdx.x & 31;
    int half = lane >> 4, lid = lane & 15;
    int m0 = wave*16;
    int n = blockIdx.x*16 + lid;
    bool nok = (n < NV3);
    int nc = nok ? n : (NV3 - 1);          // clamp loads; bad lanes never store
    const float* ap = A + (long)(m0 + lid)*LAT + 2*half;
    const float* wp = Bm + (long)(2*half)*NV3 + nc;
    v8f acc = {};
    for (int kk = 0; kk < NS; kk += 16) {
        __builtin_prefetch(wp + (long)16*NV3, 0, 3);     // global_prefetch_b8
        #pragma unroll
        for (int u = 0; u < 4; ++u) {
            v2f a = *(const v2f*)(ap);
            v2f bf; bf.x = wp[0]; bf.y = wp[NV3];
            acc = __builtin_amdgcn_wmma_f32_16x16x4_f32(false, a, false, bf,
                                                        (short)0, acc, false, false);
            ap += 4; wp += (long)4*NV3;
        }
    }
    if (nok) {
        float tv = tmpl[n];
        #pragma unroll
        for (int r = 0; r < 8; ++r) vert[(long)(m0 + 8*half + r)*NV3 + n] = acc[r] + tv;
    }
}

// ---------------- K4: per-batch geometry ----------------
// NOTE: eyeball-vertex rotations in the reference are dead code for the output
// (eye masks >= 3931 never enter face_mask < 3500, and the centres are rotation-
// invariant), so only the centres + gaze math are computed.
__global__ void geom_kernel(const float* __restrict__ latent, const float* __restrict__ bmean,
                            const float* __restrict__ vert, const int* __restrict__ lmk,
                            float* __restrict__ bpar, float* __restrict__ out) {
    __shared__ float sR[9];
    __shared__ float sMisc[8];   // [0]=scale [1..3]=t [4..6]=vmean
    __shared__ float red[256*6];
    int b = blockIdx.x, t = threadIdx.x;
    const float* lat = latent + (long)b*LAT;

    // vmean = template_mean + shape_p . basis_mean  (96 threads: 3 ch x 32 groups)
    float vmacc = 0.f;
    if (t < 96) {
        int c = t % 3, s0 = t / 3;
        for (int s = s0; s < NS; s += 32) vmacc += lat[s]*bmean[s*3 + c];
    }
    red[t] = vmacc;
    __syncthreads();
    if (t < 3) {
        float sum = 0.f;
        for (int g = 0; g < 32; ++g) sum += red[g*3 + t];
        sMisc[4+t] = sum + bmean[NS*3 + t];
    }
    if (t == 0) {
        axis_angle_to_R(lat[545], lat[546], lat[547], sR);
        sMisc[0] = lat[551] + 1.0f;
        sMisc[1] = lat[548]; sMisc[2] = lat[549]; sMisc[3] = lat[550];
    }
    __syncthreads();
    float scale = sMisc[0];

    // eye-region means over transformed verts (l: 3931..4476, r: 4477..5022)
    float ls0=0,ls1=0,ls2=0, rs0=0,rs1=0,rs2=0;
    const float* vb = vert + (long)b*NV3;
    for (int idx = 3931 + t; idx < VV; idx += 256) {
        float p[3];
        transform_point(vb + idx*3, sMisc+4, sR, scale, sMisc+1, p);
        if (idx < 4477) { ls0 += p[0]; ls1 += p[1]; ls2 += p[2]; }
        else            { rs0 += p[0]; rs1 += p[1]; rs2 += p[2]; }
    }
    __syncthreads();
    red[t] = ls0; red[256+t] = ls1; red[512+t] = ls2;
    red[768+t] = rs0; red[1024+t] = rs1; red[1280+t] = rs2;
    __syncthreads();
    for (int off = 128; off > 0; off >>= 1) {
        if (t < off)
            #pragma unroll
            for (int a = 0; a < 6; ++a) red[a*256 + t] += red[a*256 + t + off];
        __syncthreads();
    }

    if (t == 0) {
        float inv = 1.0f / 546.0f;
        float lc[3] = { red[0]*inv, red[256]*inv, red[512]*inv };
        float rc[3] = { red[768]*inv, red[1024]*inv, red[1280]*inv };

        int i4[4] = { lmk[19], lmk[22], lmk[25], lmk[28] };
        int i2[2] = { lmk[14], lmk[18] };
        float m4[3] = {0,0,0}, m2[3] = {0,0,0}, p[3];
        for (int j = 0; j < 4; ++j) {
            transform_point(vb + i4[j]*3, sMisc+4, sR, scale, sMisc+1, p);
            m4[0] += p[0]*0.25f; m4[1] += p[1]*0.25f; m4[2] += p[2]*0.25f;
        }
        for (int j = 0; j < 2; ++j) {
            transform_point(vb + i2[j]*3, sMisc+4, sR, scale, sMisc+1, p);
            m2[0] += p[0]*0.5f; m2[1] += p[1]*0.5f; m2[2] += p[2]*0.5f;
        }
        float fc[3] = { 0.5f*(m4[0]+m2[0]), 0.5f*(m4[1]+m2[1]), 0.5f*(m4[2]+m2[2]) };

        // gaze vectors: g = [0,0,-1] @ R2  => -row2(R2)
        float R2l[9], R2r[9];
        axis_angle_to_R(lat[552], lat[553], 0.f, R2l);
        axis_angle_to_R(lat[554], lat[555], 0.f, R2r);
        float lg[3] = { -R2l[6], -R2l[7], -R2l[8] };
        float rg[3] = { -R2r[6], -R2r[7], -R2r[8] };
        float cr[3] = { rg[1]*lg[2]-rg[2]*lg[1], rg[2]*lg[0]-rg[0]*lg[2], rg[0]*lg[1]-rg[1]*lg[0] };
        float c1[3] = { -rg[0], -rg[1], -rg[2] };
        float rhs[3] = { rc[0]-lc[0], rc[1]-lc[1], rc[2]-lc[2] };
        float det = triple(lg, c1, cr);
        float s0 = triple(rhs, c1, cr) / det;
        float s1 = triple(lg, rhs, cr) / det;
        float gpl[3] = { lc[0]+s0*lg[0], lc[1]+s0*lg[1], lc[2]+s0*lg[2] };
        float gpr[3] = { rc[0]+s1*rg[0], rc[1]+s1*rg[1], rc[2]+s1*rg[2] };
        float gpm[3] = { 0.5f*(gpl[0]+gpr[0]), 0.5f*(gpl[1]+gpr[1]), 0.5f*(gpl[2]+gpr[2]) };
        float dx = gpl[0]-gpr[0], dy = gpl[1]-gpr[1], dz = gpl[2]-gpr[2];
        float dist = sqrtf(dx*dx + dy*dy + dz*dz);

        float* ob = out + (long)b*VOUT*3;
        #define W3(row, v0, v1, v2) { ob[(row)*3]=(v0); ob[(row)*3+1]=(v1); ob[(row)*3+2]=(v2); }
        W3(7068, lc[0], lc[1], lc[2]);
        W3(7069, rc[0], rc[1], rc[2]);
        W3(7070, fc[0], fc[1], fc[2]);
        W3(7071, gpl[0], gpl[1], gpl[2]);
        W3(7072, gpr[0], gpr[1], gpr[2]);
        W3(7073, gpm[0], gpm[1], gpm[2]);
        W3(7074, lc[0]+lg[0]*1000.f, lc[1]+lg[1]*1000.f, lc[2]+lg[2]*1000.f);
        W3(7075, rc[0]+rg[0]*1000.f, rc[1]+rg[1]*1000.f, rc[2]+rg[2]*1000.f);
        W3(7076, lg[0], lg[1], lg[2]);
        W3(7077, rg[0], rg[1], rg[2]);
        W3(7078, dist, dist, dist);
        #undef W3

        float* bp = bpar + b*32;
        bp[0]=sMisc[4]; bp[1]=sMisc[5]; bp[2]=sMisc[6];
        #pragma unroll
        for (int j = 0; j < 9; ++j) bp[3+j] = sR[j];
        bp[12] = scale; bp[13] = sMisc[1]; bp[14] = sMisc[2]; bp[15] = sMisc[3];
    }
}

// ---------------- K5: per-vertex rigid transform + camera projection ----------------
__global__ void project_kernel(const float* __restrict__ vert, const float* __restrict__ bpar,
                               const float* __restrict__ cam, float* __restrict__ out) {
    long i = (long)blockIdx.x*blockDim.x + threadIdx.x;
    if (i >= (long)BB*VM) return;
    int b = (int)(i / VM), v = (int)(i % VM);
    const float* bp = bpar + b*32;
    float p[3];
    transform_point(vert + (long)b*NV3 + v*3, bp, bp+3, bp[12], bp+13, p);
    float* ob = out + ((long)b*VOUT + v)*3;
    ob[0] = p[0]; ob[1] = p[1]; ob[2] = p[2];
    const float* C = cam + b*12;
    float q0 = C[0]*p[0] + C[1]*p[1] + C[2]*p[2]  + C[3];
    float q1 = C[4]*p[0] + C[5]*p[1] + C[6]*p[2]  + C[7];
    float q2 = C[8]*p[0] + C[9]*p[1] + C[10]*p[2] + C[11];
    float zs = (q2 >= 0.f ? 1.f : -1.f) * fmaxf(fabsf(q2), 0.001f);
    float* oi = out + ((long)b*VOUT + VM + v)*3;
    oi[0] = q0/zs; oi[1] = q1/zs; oi[2] = q2;
}

// ---------------- K6: fl3d landmark gather ----------------
__global__ void fl3d_kernel(const int* __restrict__ ml, float* __restrict__ out) {
    int i = blockIdx.x*blockDim.x + threadIdx.x;
    if (i >= BB*68) return;
    int b = i / 68, j = i % 68;
    int v = ml[j];
    const float* src = out + ((long)b*VOUT + v)*3;
    float* dst = out + ((long)b*VOUT + 7000 + j)*3;
    dst[0] = src[0]; dst[1] = src[1]; dst[2] = src[2];
}

// ---------------- launch ----------------
extern "C" void kernel_launch(void* const* d_in, const int* in_sizes, int n_in,
                              void* d_out, int out_size, void* d_ws, size_t ws_size,
                              hipStream_t stream) {
    const float* x     = (const float*)d_in[0];
    const float* encW  = (const float*)d_in[1];
    const float* encB  = (const float*)d_in[2];
    const float* tmpl  = (const float*)d_in[3];
    const float* basis = (const float*)d_in[4];
    const float* cam   = (const float*)d_in[5];
    const int*   lmk   = (const int*)d_in[6];
    const int*   mlmk  = (const int*)d_in[7];
    float* out = (float*)d_out;

    // workspace carve-up (floats): partial | latent | bmean | vert_raw | bpar  (~13.2 MB)
    float* ws       = (float*)d_ws;
    float* wsPart   = ws;                              // KSPLIT*BB*NP = 2,293,760
    float* wsLatent = wsPart   + (long)KSPLIT*BB*NP;   // BB*LAT = 35,584
    float* wsBmean  = wsLatent + (long)BB*LAT;         // 400*3 + 3 -> pad 1216
    float* wsVert   = wsBmean  + 1216;                 // BB*NV3 = 964,416
    float* wsBpar   = wsVert   + (long)BB*NV3;         // BB*32

    means_kernel  <<<NS + 1, 256, 0, stream>>>(basis, tmpl, wsBmean);
    gemm1_wmma    <<<dim3((LAT + 15)/16, KSPLIT), 128, 0, stream>>>(x, encW, wsPart);
    reduce_latent <<<(BB*LAT + 255)/256, 256, 0, stream>>>(wsPart, encB, wsLatent);
    gemm2_wmma    <<<(NV3 + 15)/16, 128, 0, stream>>>(wsLatent, basis, tmpl, wsVert);
    geom_kernel   <<<BB, 256, 0, stream>>>(wsLatent, wsBmean, wsVert, lmk, wsBpar, out);
    project_kernel<<<((long)BB*VM + 255)/256, 256, 0, stream>>>(wsVert, wsBpar, cam, out);
    fl3d_kernel   <<<(BB*68 + 255)/256, 256, 0, stream>>>(mlmk, out);
}